// BiTemperedLogisticLoss_58377195487731
// MI455X (gfx1250) — compile-verified
//
#include <hip/hip_runtime.h>
#include <stdint.h>

// Bi-tempered logistic loss, T1=0.8, T2=1.4, label_smooth=0.1, 5 fixed-point iters.
// One workgroup per row. Row (32KB) staged global->LDS via CDNA5 async DMA
// (GLOBAL_LOAD_ASYNC_TO_LDS_B128 / s_wait_asynccnt), then held in registers
// (32 f32/thread) for the 6 reduction passes + loss pass, so HBM is touched once.

#define BLOCK  256
#define NWAVES (BLOCK / 32)
#define CCLS   8192
#define PER    (CCLS / BLOCK)   // 32 floats per thread
#define PER4   (PER / 4)        // 8 float4 per thread

__device__ __forceinline__ float fast_pow(float x, float p) {
    // x > 0 here. v_log_f32 is log2, v_exp_f32 is exp2.
    return __builtin_amdgcn_exp2f(p * __builtin_amdgcn_logf(x));
}

__device__ __forceinline__ float wave_sum(float v) {
#pragma unroll
    for (int o = 16; o > 0; o >>= 1) v += __shfl_xor(v, o, 32);
    return v;
}

__device__ __forceinline__ float wave_max(float v) {
#pragma unroll
    for (int o = 16; o > 0; o >>= 1) v = fmaxf(v, __shfl_xor(v, o, 32));
    return v;
}

__device__ __forceinline__ float block_sum(float v, float* red) {
    v = wave_sum(v);
    const int lane = threadIdx.x & 31, w = threadIdx.x >> 5;
    if (lane == 0) red[w] = v;
    __syncthreads();
    float s = 0.0f;
#pragma unroll
    for (int i = 0; i < NWAVES; ++i) s += red[i];   // same order in every thread
    __syncthreads();
    return s;
}

__device__ __forceinline__ float block_max(float v, float* red) {
    v = wave_max(v);
    const int lane = threadIdx.x & 31, w = threadIdx.x >> 5;
    if (lane == 0) red[w] = v;
    __syncthreads();
    float s = -__builtin_inff();
#pragma unroll
    for (int i = 0; i < NWAVES; ++i) s = fmaxf(s, red[i]);
    __syncthreads();
    return s;
}

__global__ void __launch_bounds__(BLOCK)
BiTemperedLogisticLoss_row_kernel(const float* __restrict__ act,
                                  const int* __restrict__ labels,
                                  float* __restrict__ rowloss) {
    __shared__ float lds_row[CCLS];
    __shared__ float red[NWAVES];

    const int row = blockIdx.x;
    const int t   = threadIdx.x;
    const float4* __restrict__ gsrc =
        reinterpret_cast<const float4*>(act + (size_t)row * CCLS);

    // ---- async global -> LDS DMA of this row (ASYNCcnt path) ----
#pragma unroll
    for (int k = 0; k < PER4; ++k) {
        const float4* gp = gsrc + (t + k * BLOCK);
        // flat shared pointer: low 32 bits are the LDS byte address
        unsigned lds_b = (unsigned)(uintptr_t)(&lds_row[4 * (t + k * BLOCK)]);
        asm volatile("global_load_async_to_lds_b128 %0, %1, off"
                     :: "v"(lds_b), "v"(gp)
                     : "memory");
    }
    asm volatile("s_wait_asynccnt 0x0" ::: "memory");
    __syncthreads();

    // ---- LDS -> registers once; everything after is register-resident ----
    float a[PER];
#pragma unroll
    for (int k = 0; k < PER4; ++k) {
        float4 f = reinterpret_cast<const float4*>(lds_row)[t + k * BLOCK];
        a[4 * k + 0] = f.x; a[4 * k + 1] = f.y;
        a[4 * k + 2] = f.z; a[4 * k + 3] = f.w;
    }

    // ---- mu = row max; a := a0 = x - mu ----
    float m = -__builtin_inff();
#pragma unroll
    for (int i = 0; i < PER; ++i) m = fmaxf(m, a[i]);
    m = block_max(m, red);
#pragma unroll
    for (int i = 0; i < PER; ++i) a[i] -= m;

    // ---- fixed-point normalization: 5 iters + final z ----
    // exp_t(u, 1.4) = relu(1 - 0.4u)^(-2.5) = rsqrt(s)^5,  s = 1 - 0.4u >= 1 here
    float w = 1.0f;
    float z = 0.0f;
#pragma unroll 1
    for (int it = 0; it <= 5; ++it) {
        const float coef  = -0.4f * w;
        float       local = 0.0f;
#pragma unroll
        for (int i = 0; i < PER; ++i) {
            const float s  = fmaf(a[i], coef, 1.0f);
            const float r  = __builtin_amdgcn_rsqf(s);
            const float r2 = r * r;
            local += r2 * r2 * r;                 // s^(-2.5)
        }
        z = block_sum(local, red);
        if (it < 5) w = fast_pow(z, -0.4f);       // a_{k+1} = a0 * z^(1-T2)
    }

    // norm - mu = (z^0.4 - 1)/0.4  =>  s_i = 1 - 0.4*(a0_i - d) = z^0.4 - 0.4*a0_i
    const float zp = fast_pow(z, 0.4f);

    // label-smoothing constants (two-valued one-hot)
    const float vOff   = 0.1f / (float)(CCLS - 1);
    const float vOn    = 0.9f;
    const float ltOff  = 5.0f * (fast_pow(vOff + 1e-10f, 0.2f) - 1.0f); // log_t(.,0.8)
    const float ltOn   = 5.0f * (fast_pow(vOn  + 1e-10f, 0.2f) - 1.0f);
    const float powOff = fast_pow(vOff, 1.2f) * (1.0f / 1.2f);
    const float powOn  = fast_pow(vOn,  1.2f) * (1.0f / 1.2f);
    const float constPerElem   = vOff * ltOff - powOff;                 // per "off" elem
    const float labelDeltaConst = (vOn * ltOn - vOff * ltOff) - (powOn - powOff);

    // ---- loss pass: p = s^(-2.5); p^0.2 = rsqrt(s); p^1.2 = s^(-3) = rsqrt(s)^6
    const int lbl = labels[row];
    float local = 0.0f;
#pragma unroll
    for (int k = 0; k < PER4; ++k) {
        const int base = 4 * (t + k * BLOCK);
#pragma unroll
        for (int j = 0; j < 4; ++j) {
            const float s    = fmaf(-0.4f, a[4 * k + j], zp);  // >= zp >= 1
            const float r    = __builtin_amdgcn_rsqf(s);
            const float r2   = r * r;
            const float lt_p = 5.0f * (r - 1.0f);              // log_t(p, 0.8)
            local += r2 * r2 * r2 * (1.0f / 1.2f) - vOff * lt_p;
            if (base + j == lbl)
                local += labelDeltaConst - (vOn - vOff) * lt_p;
        }
    }

    const float rl = block_sum(local, red) + (float)CCLS * constPerElem;
    if (t == 0) rowloss[row] = rl;
}

__global__ void __launch_bounds__(BLOCK)
BiTemperedLogisticLoss_final_kernel(const float* __restrict__ rowloss,
                                    float* __restrict__ out, int B) {
    __shared__ float red[NWAVES];
    float s = 0.0f;
    for (int i = threadIdx.x; i < B; i += BLOCK) s += rowloss[i];
    s = block_sum(s, red);
    if (threadIdx.x == 0) out[0] = s / (float)B;
}

extern "C" void kernel_launch(void* const* d_in, const int* in_sizes, int n_in,
                              void* d_out, int out_size, void* d_ws, size_t ws_size,
                              hipStream_t stream) {
    const float* act    = (const float*)d_in[0];
    const int*   labels = (const int*)d_in[1];
    float*       out    = (float*)d_out;
    float*       rl     = (float*)d_ws;   // B floats of scratch

    const int B = in_sizes[1];            // 8192 rows

    BiTemperedLogisticLoss_row_kernel<<<B, BLOCK, 0, stream>>>(act, labels, rl);
    BiTemperedLogisticLoss_final_kernel<<<1, BLOCK, 0, stream>>>(rl, out, B);
}